// GLMSSDLayer_85177791414426
// MI455X (gfx1250) — compile-verified
//
#include <hip/hip_runtime.h>
#include <math.h>

// Problem constants (match reference)
#define K4 4
#define NN 4096
#define DD 256
#define SS 32
#define ND (NN*DD)   // 1048576
// Output layout: [H_out (N*D)] [orth_loss (1)] [lap_smooth (1)] [w (4)]

typedef __attribute__((ext_vector_type(2))) float v2f;
typedef __attribute__((ext_vector_type(8))) float v8f;

// fp32 WMMA: D(16x16) = A(16x4) * B(4x16) + C ; 8-arg VOP3P form
__device__ __forceinline__ v8f wmma4(v2f a, v2f b, v8f c) {
    return __builtin_amdgcn_wmma_f32_16x16x4_f32(
        /*neg_a=*/false, a, /*neg_b=*/false, b,
        /*c_mod=*/(short)0, c, /*reuse_a=*/false, /*reuse_b=*/false);
}

// ---------------- workspace layout (floats) ----------------
#define WS_W    0
#define WS_G    16
#define WS_MINV 4112
#define WS_ZT   8208
#define WS_MZT  532496
#define WS_OKW  1056784
#define WS_HAGG 1581072
// total 2629648 floats = ~10.5 MB

// -------- init: zero reduction targets, softmax(hop_weights) --------
__global__ void glm_init(const float* __restrict__ hw, float* __restrict__ out,
                         float* __restrict__ ws) {
    int t = threadIdx.x + blockIdx.x * blockDim.x;
    if (t < K4 * SS * SS) ws[WS_G + t] = 0.0f;
    if (t == 0) {
        float m = hw[0];
        for (int i = 1; i < K4; i++) m = fmaxf(m, hw[i]);
        float e[K4], s = 0.0f;
        for (int i = 0; i < K4; i++) { e[i] = __expf(hw[i] - m); s += e[i]; }
        for (int i = 0; i < K4; i++) {
            float wv = e[i] / s;
            ws[WS_W + i] = wv;
            out[ND + 2 + i] = wv;
        }
        out[ND]     = 0.0f;  // orth_loss accumulator
        out[ND + 1] = 0.0f;  // lap_smooth accumulator
    }
}

// -------- Zt[k] = hops[k] (NxD) @ U[k] (DxS)  -> (NxS), fp32 WMMA --------
__global__ void glm_zt(const float* __restrict__ hops, const float* __restrict__ U,
                       float* __restrict__ Zt) {
    const int lane = threadIdx.x & 31, half = lane >> 4, lr = lane & 15;
    const int base = 2 * half;
    const int nb = blockIdx.x * 16, sb = blockIdx.y * 16, k = blockIdx.z;
    const float* __restrict__ H  = hops + (size_t)k * NN * DD;
    const float* __restrict__ Uk = U    + (size_t)k * DD * SS;
    v8f c = {};
    for (int t = 0; t < DD / 4; t++) {
        int kc = 4 * t + base;
        v2f a, b;
        a.x = H[(size_t)(nb + lr) * DD + kc];
        a.y = H[(size_t)(nb + lr) * DD + kc + 1];
        b.x = Uk[(size_t)kc * SS + sb + lr];
        b.y = Uk[(size_t)(kc + 1) * SS + sb + lr];
        c = wmma4(a, b, c);
    }
    float* __restrict__ Z = Zt + (size_t)k * NN * SS;
    for (int g = 0; g < 8; g++)
        Z[(size_t)(nb + g + 8 * half) * SS + sb + lr] = c[g];
}

// -------- Gram: G[k] = Zt[k]^T @ Zt[k]  (32x32) --------
__global__ void glm_gram(const float* __restrict__ Zt, float* __restrict__ G) {
    const int k = blockIdx.y, chunk = blockIdx.x;         // 32 chunks of 128 rows
    const int s1 = threadIdx.x >> 5, s2 = threadIdx.x & 31;
    const float* __restrict__ Z = Zt + (size_t)k * NN * SS;
    float acc = 0.0f;
    const int n0 = chunk * 128;
    for (int n = n0; n < n0 + 128; n++)
        acc += Z[(size_t)n * SS + s1] * Z[(size_t)n * SS + s2];
    atomicAdd(&G[(size_t)k * SS * SS + s1 * SS + s2], acc);
}

// -------- Minv = (I + coeff*G)^-1 via Gauss-Jordan (SPD, no pivoting) --------
// Single wave (32 threads); lockstep makes the row-op read/write safe.
__global__ void glm_minv(const float* __restrict__ G, float* __restrict__ Minv) {
    __shared__ float M[SS][SS + 1];
    __shared__ float X[SS][SS + 1];
    const int j = threadIdx.x, k = blockIdx.x;
    const float coeff = (float)SS / ((float)NN * 0.5f * 0.5f);  // S/(N*EPS^2)
    for (int i = 0; i < SS; i++) {
        M[i][j] = coeff * G[(size_t)k * SS * SS + i * SS + j] + (i == j ? 1.0f : 0.0f);
        X[i][j] = (i == j) ? 1.0f : 0.0f;
    }
    __syncthreads();
    for (int i = 0; i < SS; i++) {
        float inv = 1.0f / M[i][i];
        M[i][j] *= inv;
        X[i][j] *= inv;
        __syncthreads();
        for (int r = 0; r < SS; r++) {
            if (r != i) {
                float f = M[r][i];        // all lanes read before any lane writes
                M[r][j] -= f * M[i][j];
                X[r][j] -= f * X[i][j];
            }
        }
        __syncthreads();
    }
    for (int i = 0; i < SS; i++)
        Minv[(size_t)k * SS * SS + i * SS + j] = X[i][j];
}

// -------- MZt[k] = Zt[k] (NxS) @ Minv[k] (SxS) --------
__global__ void glm_mzt(const float* __restrict__ Zt, const float* __restrict__ Minv,
                        float* __restrict__ MZt) {
    const int lane = threadIdx.x & 31, half = lane >> 4, lr = lane & 15;
    const int base = 2 * half;
    const int nb = blockIdx.x * 16, sb = blockIdx.y * 16, k = blockIdx.z;
    const float* __restrict__ Z  = Zt   + (size_t)k * NN * SS;
    const float* __restrict__ Mi = Minv + (size_t)k * SS * SS;
    v8f c = {};
    for (int t = 0; t < SS / 4; t++) {
        int kc = 4 * t + base;
        v2f a, b;
        a.x = Z[(size_t)(nb + lr) * SS + kc];
        a.y = Z[(size_t)(nb + lr) * SS + kc + 1];
        b.x = Mi[kc * SS + sb + lr];        // Minv is symmetric
        b.y = Mi[(kc + 1) * SS + sb + lr];
        c = wmma4(a, b, c);
    }
    float* __restrict__ O = MZt + (size_t)k * NN * SS;
    for (int g = 0; g < 8; g++)
        O[(size_t)(nb + g + 8 * half) * SS + sb + lr] = c[g];
}

// -------- Flash attention: Okw = w[k] * softmax(Zt@MZt^T + mask) @ MZt --------
__global__ void glm_attn(const float* __restrict__ Zt, const float* __restrict__ MZt,
                         const float* __restrict__ adj, const float* __restrict__ wv,
                         float* __restrict__ Okw) {
    __shared__ float ptile[16][20];
    const int lane = threadIdx.x & 31, half = lane >> 4, lr = lane & 15;
    const int base = 2 * half;
    const int qb = blockIdx.x * 16, k = blockIdx.z;
    const float* __restrict__ Z = Zt  + (size_t)k * NN * SS;
    const float* __restrict__ V = MZt + (size_t)k * NN * SS;   // K == V == MZt

    v2f qa[8];
    for (int t = 0; t < 8; t++) {
        int kc = 4 * t + base;
        qa[t].x = Z[(size_t)(qb + lr) * SS + kc];
        qa[t].y = Z[(size_t)(qb + lr) * SS + kc + 1];
    }
    float m[8], ssum[8];
    v8f o0 = {}, o1 = {};
    for (int g = 0; g < 8; g++) { m[g] = -3.0e38f; ssum[g] = 0.0f; }

    for (int jt = 0; jt < NN / 16; jt++) {
        const int kb = jt * 16;
        // S_tile = Q @ K^T  (B[s, key] = MZt[kb+key, s] -> row reads per lane)
        v8f c = {};
        for (int t = 0; t < 8; t++) {
            int kc = 4 * t + base;
            v2f b;
            b.x = V[(size_t)(kb + lr) * SS + kc];
            b.y = V[(size_t)(kb + lr) * SS + kc + 1];
            c = wmma4(qa[t], b, c);
        }
        // mask + online softmax (rows live in 16-lane half-groups)
        float p[8];
        for (int g = 0; g < 8; g++) {
            const int row = qb + g + 8 * half;
            float aval = adj[(size_t)row * NN + kb + lr];
            float sc = c[g] + (1.0f - aval) * (-1.0e9f);
            float mx = sc;
            for (int off = 1; off < 16; off <<= 1)
                mx = fmaxf(mx, __shfl_xor(mx, off, 32));
            float nm = fmaxf(m[g], mx);
            float pv = __expf(sc - nm);
            float ps = pv;
            for (int off = 1; off < 16; off <<= 1)
                ps += __shfl_xor(ps, off, 32);
            float rsc = __expf(m[g] - nm);
            ssum[g] = ssum[g] * rsc + ps;
            m[g] = nm;
            o0[g] *= rsc;
            o1[g] *= rsc;
            p[g] = pv;
        }
        // restage P (C layout) -> LDS -> A layout
        for (int g = 0; g < 8; g++) ptile[g + 8 * half][lr] = p[g];
        asm volatile("s_wait_dscnt 0" ::: "memory");
        // O += P @ V   (two 16x16 output tiles, K = 16 keys)
        for (int t = 0; t < 4; t++) {
            int kc = 4 * t + base;
            v2f pa, b0, b1;
            pa.x = ptile[lr][kc];
            pa.y = ptile[lr][kc + 1];
            b0.x = V[(size_t)(kb + kc) * SS + lr];
            b0.y = V[(size_t)(kb + kc + 1) * SS + lr];
            b1.x = V[(size_t)(kb + kc) * SS + 16 + lr];
            b1.y = V[(size_t)(kb + kc + 1) * SS + 16 + lr];
            o0 = wmma4(pa, b0, o0);
            o1 = wmma4(pa, b1, o1);
        }
    }
    const float wk = wv[k];
    float* __restrict__ O = Okw + (size_t)k * NN * SS;
    for (int g = 0; g < 8; g++) {
        const float inv = wk / ssum[g];
        const int row = qb + g + 8 * half;
        O[(size_t)row * SS + lr]      = o0[g] * inv;
        O[(size_t)row * SS + 16 + lr] = o1[g] * inv;
    }
}

// -------- H_agg (NxD) = sum_k Okw[k] (NxS) @ U[k]^T (SxD) --------
__global__ void glm_hagg(const float* __restrict__ Okw, const float* __restrict__ U,
                         float* __restrict__ Hagg) {
    const int lane = threadIdx.x & 31, half = lane >> 4, lr = lane & 15;
    const int base = 2 * half;
    const int nb = blockIdx.x * 16, db = blockIdx.y * 16;
    v8f c = {};
    for (int k = 0; k < K4; k++) {
        const float* __restrict__ O  = Okw + (size_t)k * NN * SS;
        const float* __restrict__ Uk = U   + (size_t)k * DD * SS;
        for (int t = 0; t < SS / 4; t++) {
            int kc = 4 * t + base;
            v2f a, b;
            a.x = O[(size_t)(nb + lr) * SS + kc];
            a.y = O[(size_t)(nb + lr) * SS + kc + 1];
            b.x = Uk[(size_t)(db + lr) * SS + kc];      // B[s,d] = Uk[d,s]
            b.y = Uk[(size_t)(db + lr) * SS + kc + 1];
            c = wmma4(a, b, c);
        }
    }
    for (int g = 0; g < 8; g++)
        Hagg[(size_t)(nb + g + 8 * half) * DD + db + lr] = c[g];
}

// -------- H_out = softthresh(hops0 + ETA*Hagg - ETA*lam*(L@Hagg)) --------
// 32 n-rows x 64 d-cols per wave: 8 WMMAs per k-step (2 n-tiles x 4 d-tiles); each B tile feeds 2
__global__ void glm_hout(const float* __restrict__ L, const float* __restrict__ Hagg,
                         const float* __restrict__ hops, const float* __restrict__ thr,
                         const float* __restrict__ lam, float* __restrict__ out) {
    const int lane = threadIdx.x & 31, half = lane >> 4, lr = lane & 15;
    const int base = 2 * half;
    const int nb = blockIdx.x * 32, db0 = blockIdx.y * 64;
    const float* __restrict__ Lr0 = L + (size_t)(nb + lr) * NN;
    const float* __restrict__ Lr1 = L + (size_t)(nb + 16 + lr) * NN;
    v8f c0[4] = {};
    v8f c1[4] = {};
    for (int t = 0; t < NN / 4; t++) {
        const int kc = 4 * t + base;
        v2f a0, a1;
        a0.x = Lr0[kc]; a0.y = Lr0[kc + 1];
        a1.x = Lr1[kc]; a1.y = Lr1[kc + 1];
        const float* __restrict__ H0 = Hagg + (size_t)kc * DD + db0 + lr;
        const float* __restrict__ H1 = Hagg + (size_t)(kc + 1) * DD + db0 + lr;
        for (int q = 0; q < 4; q++) {
            v2f b;
            b.x = H0[16 * q];
            b.y = H1[16 * q];
            c0[q] = wmma4(a0, b, c0[q]);
            c1[q] = wmma4(a1, b, c1[q]);
        }
    }
    const float lv = lam[0];
    for (int q = 0; q < 4; q++) {
        for (int g = 0; g < 8; g++) {
            const int d = db0 + 16 * q + lr;
            const float th = thr[d];
            {
                const int n = nb + g + 8 * half;
                float hh = hops[(size_t)n * DD + d]
                         + 0.5f * Hagg[(size_t)n * DD + d]
                         - 0.5f * lv * c0[q][g];
                float ab = fabsf(hh) - th;
                out[(size_t)n * DD + d] = (ab > 0.0f) ? copysignf(ab, hh) : 0.0f;
            }
            {
                const int n = nb + 16 + g + 8 * half;
                float hh = hops[(size_t)n * DD + d]
                         + 0.5f * Hagg[(size_t)n * DD + d]
                         - 0.5f * lv * c1[q][g];
                float ab = fabsf(hh) - th;
                out[(size_t)n * DD + d] = (ab > 0.0f) ? copysignf(ab, hh) : 0.0f;
            }
        }
    }
}

// -------- lap_smooth = sum(H_out * (L @ H_out)) -------- (same 32x64 tiling)
__global__ void glm_smooth(const float* __restrict__ L, const float* __restrict__ Hout,
                           float* __restrict__ out) {
    const int lane = threadIdx.x & 31, half = lane >> 4, lr = lane & 15;
    const int base = 2 * half;
    const int nb = blockIdx.x * 32, db0 = blockIdx.y * 64;
    const float* __restrict__ Lr0 = L + (size_t)(nb + lr) * NN;
    const float* __restrict__ Lr1 = L + (size_t)(nb + 16 + lr) * NN;
    v8f c0[4] = {};
    v8f c1[4] = {};
    for (int t = 0; t < NN / 4; t++) {
        const int kc = 4 * t + base;
        v2f a0, a1;
        a0.x = Lr0[kc]; a0.y = Lr0[kc + 1];
        a1.x = Lr1[kc]; a1.y = Lr1[kc + 1];
        const float* __restrict__ H0 = Hout + (size_t)kc * DD + db0 + lr;
        const float* __restrict__ H1 = Hout + (size_t)(kc + 1) * DD + db0 + lr;
        for (int q = 0; q < 4; q++) {
            v2f b;
            b.x = H0[16 * q];
            b.y = H1[16 * q];
            c0[q] = wmma4(a0, b, c0[q]);
            c1[q] = wmma4(a1, b, c1[q]);
        }
    }
    float acc = 0.0f;
    for (int q = 0; q < 4; q++) {
        for (int g = 0; g < 8; g++) {
            const int d = db0 + 16 * q + lr;
            const int n0 = nb + g + 8 * half;
            const int n1 = nb + 16 + g + 8 * half;
            acc += c0[q][g] * Hout[(size_t)n0 * DD + d];
            acc += c1[q][g] * Hout[(size_t)n1 * DD + d];
        }
    }
    for (int off = 1; off < 32; off <<= 1) acc += __shfl_xor(acc, off, 32);
    if (lane == 0) atomicAdd(&out[ND + 1], acc);
}

// -------- orth_loss over the 10 (k<=l) pairs of U --------
__global__ void glm_orth(const float* __restrict__ U, float* __restrict__ out) {
    __shared__ float red[32];
    const int pk[10] = {0, 0, 0, 0, 1, 1, 1, 2, 2, 3};
    const int pl[10] = {0, 1, 2, 3, 1, 2, 3, 2, 3, 3};
    const int p = blockIdx.x, k = pk[p], l = pl[p];
    const int s1 = threadIdx.x >> 5, s2 = threadIdx.x & 31;
    const float* __restrict__ Uk = U + (size_t)k * DD * SS;
    const float* __restrict__ Ul = U + (size_t)l * DD * SS;
    float dot = 0.0f;
    for (int d = 0; d < DD; d++)
        dot += Uk[(size_t)d * SS + s1] * Ul[(size_t)d * SS + s2];
    if (k == l && s1 == s2) dot -= 1.0f;
    float v = dot * dot;
    for (int off = 1; off < 32; off <<= 1) v += __shfl_xor(v, off, 32);
    if ((threadIdx.x & 31) == 0) red[threadIdx.x >> 5] = v;
    __syncthreads();
    if (threadIdx.x < 32) {
        float x = red[threadIdx.x];
        for (int off = 1; off < 32; off <<= 1) x += __shfl_xor(x, off, 32);
        if (threadIdx.x == 0) atomicAdd(&out[ND], x);
    }
}

extern "C" void kernel_launch(void* const* d_in, const int* in_sizes, int n_in,
                              void* d_out, int out_size, void* d_ws, size_t ws_size,
                              hipStream_t stream) {
    const float* hops = (const float*)d_in[0];   // (K,N,D)
    const float* adj  = (const float*)d_in[1];   // (N,N)
    const float* L    = (const float*)d_in[2];   // (N,N)
    const float* U    = (const float*)d_in[3];   // (K,D,S)
    const float* hw   = (const float*)d_in[4];   // (K,)
    const float* thr  = (const float*)d_in[5];   // (D,)
    const float* lam  = (const float*)d_in[6];   // scalar

    float* out = (float*)d_out;
    float* ws  = (float*)d_ws;   // needs ~10.5 MB

    float* wsW    = ws + WS_W;
    float* wsG    = ws + WS_G;
    float* wsMinv = ws + WS_MINV;
    float* wsZt   = ws + WS_ZT;
    float* wsMZt  = ws + WS_MZT;
    float* wsOkw  = ws + WS_OKW;
    float* wsHagg = ws + WS_HAGG;

    glm_init<<<dim3((K4 * SS * SS + 255) / 256), 256, 0, stream>>>(hw, out, ws);
    glm_zt  <<<dim3(NN / 16, SS / 16, K4), 32, 0, stream>>>(hops, U, wsZt);
    glm_gram<<<dim3(32, K4), 1024, 0, stream>>>(wsZt, wsG);
    glm_minv<<<dim3(K4), 32, 0, stream>>>(wsG, wsMinv);
    glm_mzt <<<dim3(NN / 16, SS / 16, K4), 32, 0, stream>>>(wsZt, wsMinv, wsMZt);
    glm_attn<<<dim3(NN / 16, 1, K4), 32, 0, stream>>>(wsZt, wsMZt, adj, wsW, wsOkw);
    glm_hagg<<<dim3(NN / 16, DD / 16), 32, 0, stream>>>(wsOkw, U, wsHagg);
    glm_hout<<<dim3(NN / 32, DD / 64), 32, 0, stream>>>(L, wsHagg, hops, thr, lam, out);
    glm_smooth<<<dim3(NN / 32, DD / 64), 32, 0, stream>>>(L, out, out);
    glm_orth<<<dim3(10), 1024, 0, stream>>>(U, out);
}